// PHVClassifier_15049565405193
// MI455X (gfx1250) — compile-verified
//
#include <hip/hip_runtime.h>
#include <hip/hip_bf16.h>

typedef __attribute__((ext_vector_type(2))) float v2f;
typedef __attribute__((ext_vector_type(8))) float v8f;

#define HID 32
#define H1  128

// ---------------- degree / normalization ----------------
__global__ void deg_kernel(const int* __restrict__ dst, float* __restrict__ deg, int E) {
  int t = blockIdx.x * blockDim.x + threadIdx.x;
  if (t < E) atomicAdd(&deg[dst[t]], 1.0f);
}

__global__ void dinv_kernel(float* __restrict__ degdinv, int N) {
  int t = blockIdx.x * blockDim.x + threadIdx.x;
  if (t < N) degdinv[t] = rsqrtf(degdinv[t] + 1.0f);  // +1 = self loop, always > 0
}

// ---------------- fp32 WMMA GEMM: C[M,N] = A[M,K] @ B[K,N] ----------------
// One wave per 16x16 output tile; K stepped by 4 via V_WMMA_F32_16X16X4_F32.
__global__ __launch_bounds__(256)
void gemm_wmma_f32(const float* __restrict__ A, const float* __restrict__ B,
                   float* __restrict__ C, int M, int N, int K) {
  int wave = (int)((blockIdx.x * (long long)blockDim.x + threadIdx.x) >> 5);
  int lane = threadIdx.x & 31;
  int ntiles = N >> 4;
  int mtiles = (M + 15) >> 4;
  int tm = wave / ntiles;
  int tn = wave - tm * ntiles;
  if (tm >= mtiles) return;                 // wave-uniform exit: EXEC stays full
  int mrow = tm * 16 + (lane & 15);
  if (mrow > M - 1) mrow = M - 1;           // clamp keeps EXEC all-ones for WMMA
  int ncol = tn * 16 + (lane & 15);
  int kb = (lane >> 4) * 2;                 // K pair base: 0 (lanes 0-15) / 2 (lanes 16-31)
  v8f acc = {};
  for (int k0 = 0; k0 < K; k0 += 4) {
    v2f a = *(const v2f*)(A + (size_t)mrow * K + k0 + kb);   // A[m][k0+kb], A[m][k0+kb+1]
    v2f b;
    b.x = B[(size_t)(k0 + kb) * N + ncol];
    b.y = B[(size_t)(k0 + kb + 1) * N + ncol];
    acc = __builtin_amdgcn_wmma_f32_16x16x4_f32(false, a, false, b, (short)0, acc,
                                                false, false);
  }
  int rbase = tm * 16 + ((lane >> 4) ? 8 : 0);  // VGPR r holds row rbase + r
#pragma unroll
  for (int r = 0; r < 8; ++r) {
    int row = rbase + r;
    if (row < M) C[(size_t)row * N + ncol] = acc[r];
  }
}

// ---------------- edge-normalized scatter add ----------------
template <int D>
__global__ __launch_bounds__(256)
void scatter_kernel(const float* __restrict__ xl, const int* __restrict__ src,
                    const int* __restrict__ dst, const float* __restrict__ dinv,
                    float* __restrict__ agg, int E) {
  const int LPE = D / 4;
  long long t = (long long)blockIdx.x * blockDim.x + threadIdx.x;
  int e = (int)(t / LPE);
  if (e >= E) return;
  int c = (int)(t % LPE) * 4;
  int s = src[e], d = dst[e];
  float nrm = dinv[s] * dinv[d];
  const float4 v = *(const float4*)(xl + (size_t)s * D + c);
  float* p = agg + (size_t)d * D + c;
  atomicAdd(p + 0, nrm * v.x);
  atomicAdd(p + 1, nrm * v.y);
  atomicAdd(p + 2, nrm * v.z);
  atomicAdd(p + 3, nrm * v.w);
}

// ---------------- self-loop + bias + relu, in place on agg ----------------
template <int D>
__global__ __launch_bounds__(256)
void fuse_kernel(float* __restrict__ agg, const float* __restrict__ xl,
                 const float* __restrict__ dinv, const float* __restrict__ bias, int N) {
  const int LPN = D / 4;
  long long t = (long long)blockIdx.x * blockDim.x + threadIdx.x;
  int i = (int)(t / LPN);
  if (i >= N) return;
  int c = (int)(t % LPN) * 4;
  float di = dinv[i], sl = di * di;
  float4 a = *(const float4*)(agg + (size_t)i * D + c);
  const float4 v = *(const float4*)(xl + (size_t)i * D + c);
  a.x = fmaxf(a.x + sl * v.x + bias[c + 0], 0.f);
  a.y = fmaxf(a.y + sl * v.y + bias[c + 1], 0.f);
  a.z = fmaxf(a.z + sl * v.z + bias[c + 2], 0.f);
  a.w = fmaxf(a.w + sl * v.w + bias[c + 3], 0.f);
  *(float4*)(agg + (size_t)i * D + c) = a;
}

// ---------------- edge MLP: one wave = 16 edges ----------------
// feat[16,64] = [h[src] | h[dst]]; t = relu(feat@Wm1 + bm1) via 2x(16 WMMA);
// out = t@Wm2 + bm2 folded via half-wave shuffle reduction.
__global__ __launch_bounds__(256)
void edge_mlp_wmma(const float* __restrict__ h, const int* __restrict__ p0,
                   const int* __restrict__ p1, const int* __restrict__ q0,
                   const int* __restrict__ q1, const float* __restrict__ Wm1,
                   const float* __restrict__ bm1, const float* __restrict__ Wm2,
                   const float* __restrict__ bm2, float* __restrict__ out,
                   int NP, int Etot) {
  int wave = (int)((blockIdx.x * (long long)blockDim.x + threadIdx.x) >> 5);
  int lane = threadIdx.x & 31;
  int e0 = wave * 16;
  if (e0 >= Etot) return;                   // wave-uniform
  int m = lane & 15;
  int e = e0 + m;
  if (e > Etot - 1) e = Etot - 1;           // clamp keeps EXEC full
  int s = (e < NP) ? p0[e] : q0[e - NP];
  int d = (e < NP) ? p1[e] : q1[e - NP];
  const float* hs = h + (size_t)s * HID;
  const float* hd = h + (size_t)d * HID;
  int kb = (lane >> 4) * 2;
  v8f acc0 = {}, acc1 = {};
#pragma unroll
  for (int ks = 0; ks < 16; ++ks) {         // K = 64 in steps of 4
    int j = ks * 4 + kb;
    const float* ap = (j < HID) ? (hs + j) : (hd + (j - HID));
    v2f a; a.x = ap[0]; a.y = ap[1];
    v2f b0, b1;
    b0.x = Wm1[(size_t)j * HID + m];
    b0.y = Wm1[(size_t)(j + 1) * HID + m];
    b1.x = Wm1[(size_t)j * HID + 16 + m];
    b1.y = Wm1[(size_t)(j + 1) * HID + 16 + m];
    acc0 = __builtin_amdgcn_wmma_f32_16x16x4_f32(false, a, false, b0, (short)0, acc0,
                                                 false, false);
    acc1 = __builtin_amdgcn_wmma_f32_16x16x4_f32(false, a, false, b1, (short)0, acc1,
                                                 false, false);
  }
  // acc0: n = m (0..15); acc1: n = m+16; row = r + 8*(lane>>4)
  float bA = bm1[m],        bB = bm1[m + 16];
  float wa0 = Wm2[m * 2],   wa1 = Wm2[m * 2 + 1];
  float wb0 = Wm2[(m + 16) * 2], wb1 = Wm2[(m + 16) * 2 + 1];
  float p[8][2];
#pragma unroll
  for (int r = 0; r < 8; ++r) {
    float t0 = fmaxf(acc0[r] + bA, 0.f);
    float t1 = fmaxf(acc1[r] + bB, 0.f);
    p[r][0] = t0 * wa0 + t1 * wb0;
    p[r][1] = t0 * wa1 + t1 * wb1;
  }
#pragma unroll
  for (int off = 8; off >= 1; off >>= 1) {
#pragma unroll
    for (int r = 0; r < 8; ++r) {
      p[r][0] += __shfl_xor(p[r][0], off, 16);
      p[r][1] += __shfl_xor(p[r][1], off, 16);
    }
  }
  if (m == 0) {
    int rbase = e0 + ((lane >> 4) ? 8 : 0);
    float c0 = bm2[0], c1 = bm2[1];
#pragma unroll
    for (int r = 0; r < 8; ++r) {
      int row = rbase + r;
      if (row < Etot) {
        out[(size_t)row * 2 + 0] = p[r][0] + c0;
        out[(size_t)row * 2 + 1] = p[r][1] + c1;
      }
    }
  }
}

// ---------------- launch ----------------
extern "C" void kernel_launch(void* const* d_in, const int* in_sizes, int n_in,
                              void* d_out, int out_size, void* d_ws, size_t ws_size,
                              hipStream_t stream) {
  const float* x   = (const float*)d_in[0];
  const int*   pe  = (const int*)d_in[1];
  const int*   neg = (const int*)d_in[2];
  const float* W1  = (const float*)d_in[3];
  const float* b1  = (const float*)d_in[4];
  const float* W2  = (const float*)d_in[5];
  const float* b2  = (const float*)d_in[6];
  const float* Wm1 = (const float*)d_in[7];
  const float* bm1 = (const float*)d_in[8];
  const float* Wm2 = (const float*)d_in[9];
  const float* bm2 = (const float*)d_in[10];
  float* out = (float*)d_out;

  const int N  = in_sizes[0] / HID;
  const int NP = in_sizes[1] / 2;
  const int NN = in_sizes[2] / 2;
  const int ET = NP + NN;
  const int* p0 = pe;        const int* p1 = pe + NP;
  const int* q0 = neg;       const int* q1 = neg + NN;

  char* ws = (char*)d_ws;
  size_t off = 0;
  auto wsalloc = [&](size_t bytes) -> char* {
    char* p = ws + off;
    off += (bytes + 255) & ~(size_t)255;
    return p;
  };
  float* dinv = (float*)wsalloc((size_t)N * 4);           // deg -> dinv in place
  float* xl1  = (float*)wsalloc((size_t)N * H1 * 4);
  float* h1   = (float*)wsalloc((size_t)N * H1 * 4);      // agg1 -> h1 in place
  float* xl2  = (float*)wsalloc((size_t)N * HID * 4);
  float* h2   = (float*)wsalloc((size_t)N * HID * 4);     // agg2 -> h in place
  (void)ws_size; (void)n_in; (void)out_size;

  const int B = 256;

  // degree + normalization
  hipMemsetAsync(dinv, 0, (size_t)N * 4, stream);
  deg_kernel<<<(NP + B - 1) / B, B, 0, stream>>>(p1, dinv, NP);
  dinv_kernel<<<(N + B - 1) / B, B, 0, stream>>>(dinv, N);

  // layer 1: xl1 = x @ W1
  {
    long long thr = (long long)((N + 15) / 16) * (H1 / 16) * 32;
    gemm_wmma_f32<<<(unsigned)((thr + B - 1) / B), B, 0, stream>>>(x, W1, xl1, N, H1, HID);
  }
  hipMemsetAsync(h1, 0, (size_t)N * H1 * 4, stream);
  {
    long long thr = (long long)NP * (H1 / 4);
    scatter_kernel<H1><<<(unsigned)((thr + B - 1) / B), B, 0, stream>>>(xl1, p0, p1, dinv, h1, NP);
    long long thr2 = (long long)N * (H1 / 4);
    fuse_kernel<H1><<<(unsigned)((thr2 + B - 1) / B), B, 0, stream>>>(h1, xl1, dinv, b1, N);
  }

  // layer 2: xl2 = h1 @ W2
  {
    long long thr = (long long)((N + 15) / 16) * (HID / 16) * 32;
    gemm_wmma_f32<<<(unsigned)((thr + B - 1) / B), B, 0, stream>>>(h1, W2, xl2, N, HID, H1);
  }
  hipMemsetAsync(h2, 0, (size_t)N * HID * 4, stream);
  {
    long long thr = (long long)NP * (HID / 4);
    scatter_kernel<HID><<<(unsigned)((thr + B - 1) / B), B, 0, stream>>>(xl2, p0, p1, dinv, h2, NP);
    long long thr2 = (long long)N * (HID / 4);
    fuse_kernel<HID><<<(unsigned)((thr2 + B - 1) / B), B, 0, stream>>>(h2, xl2, dinv, b2, N);
  }

  // edge MLP over pos+neg edges
  {
    long long thr = (long long)((ET + 15) / 16) * 32;
    edge_mlp_wmma<<<(unsigned)((thr + B - 1) / B), B, 0, stream>>>(h2, p0, p1, q0, q1,
                                                                   Wm1, bm1, Wm2, bm2,
                                                                   out, NP, ET);
  }
}